// Unpool_9139690406277
// MI455X (gfx1250) — compile-verified
//
#include <hip/hip_runtime.h>
#include <hip/hip_bf16.h>
#include <stdint.h>

// ---------------------------------------------------------------------------
// Unpool forward on MI455X (gfx1250).
//
// Reference output = (g passthrough, new_h = scatter(h, idx) into zeros).
// Pure memory-bound: ~3.26 GB of traffic, 0 FLOPs -> ~140us floor @ 23.3TB/s.
// No matmul exists here, so WMMA is inapplicable; the CDNA5-specific feature
// that applies is the async VMEM<->LDS path (GLOBAL_LOAD_ASYNC_TO_LDS_B128 /
// GLOBAL_STORE_ASYNC_FROM_LDS_B128, ASYNCcnt), used for the bulk g copy so
// the 3.2 GB stream never occupies VGPRs.
// ---------------------------------------------------------------------------

#if defined(__has_builtin)
#define HAVE_ASYNC_BUILTINS                                                    \
  (__has_builtin(__builtin_amdgcn_global_load_async_to_lds_b128) &&            \
   __has_builtin(__builtin_amdgcn_global_store_async_from_lds_b128))
#define HAVE_WAIT_ASYNC __has_builtin(__builtin_amdgcn_s_wait_asynccnt)
#else
#define HAVE_ASYNC_BUILTINS 0
#define HAVE_WAIT_ASYNC 0
#endif

// Exact parameter types per clang's diagnostic:
//   arg0: AS(1) pointer to int __attribute__((vector_size(16)))   (global)
//   arg1: AS(3) pointer (LDS), same pointee
typedef int v4i __attribute__((__vector_size__(16)));
typedef __attribute__((address_space(1))) v4i* g_v4i_p;
typedef __attribute__((address_space(3))) v4i* l_v4i_p;

// Low 32 bits of a flat pointer into LDS == wave-relative LDS byte offset
// (CDNA5 ISA: LDS aperture maps addr[31:0] -> LDS_ADDR).
__device__ __forceinline__ unsigned lds_off32(const void* p) {
  return (unsigned)(uintptr_t)p;
}

__device__ __forceinline__ void async_wait0() {
#if HAVE_WAIT_ASYNC
  __builtin_amdgcn_s_wait_asynccnt(0);
#else
  asm volatile("s_wait_asynccnt 0" ::: "memory");
#endif
}

// global -> LDS, 16 bytes per lane, tracked by ASYNCcnt
__device__ __forceinline__ void g2lds_b128(const void* gsrc, const void* lslot) {
#if HAVE_ASYNC_BUILTINS
  __builtin_amdgcn_global_load_async_to_lds_b128(
      (g_v4i_p)(uintptr_t)gsrc, (l_v4i_p)(uintptr_t)lds_off32(lslot), 0, 0);
#else
  unsigned off = lds_off32(lslot);
  asm volatile("global_load_async_to_lds_b128 %0, %1, off"
               :: "v"(off), "v"(gsrc) : "memory");
#endif
}

// LDS -> global, 16 bytes per lane, tracked by ASYNCcnt
__device__ __forceinline__ void lds2g_b128(void* gdst, const void* lslot) {
#if HAVE_ASYNC_BUILTINS
  __builtin_amdgcn_global_store_async_from_lds_b128(
      (g_v4i_p)(uintptr_t)gdst, (l_v4i_p)(uintptr_t)lds_off32(lslot), 0, 0);
#else
  unsigned off = lds_off32(lslot);
  asm volatile("global_store_async_from_lds_b128 %0, %1, off"
               :: "v"(gdst), "v"(off) : "memory");
#endif
}

// ---------------------------------------------------------------------------
// Kernel 1: bulk copy of g (N*N floats) through the async LDS DMA path.
// 256 threads (8 wave32s), 8 outstanding b128 async loads per lane, wait,
// 8 async stores straight from LDS, wait (LDS reuse safety). MLP across
// 8 waves/block * thousands of blocks saturates HBM.
// ---------------------------------------------------------------------------
#define CPY_BLOCK 256
#define CPY_UNROLL 8

__global__ __launch_bounds__(CPY_BLOCK) void copy_g_async(
    const float4* __restrict__ src, float4* __restrict__ dst, size_t total4) {
  __shared__ float4 stage[CPY_BLOCK * CPY_UNROLL];  // 32 KB
  const unsigned tid = threadIdx.x;
  const size_t nthreads = (size_t)gridDim.x * CPY_BLOCK;
  const size_t sweep = nthreads * CPY_UNROLL;

  for (size_t i = (size_t)blockIdx.x * CPY_BLOCK + tid; i < total4; i += sweep) {
#pragma unroll
    for (int u = 0; u < CPY_UNROLL; ++u) {
      size_t j = i + (size_t)u * nthreads;           // coalesced per issue
      if (j < total4) g2lds_b128(src + j, &stage[tid + u * CPY_BLOCK]);
    }
    async_wait0();  // loads landed in LDS (LDS side of async is unordered)
#pragma unroll
    for (int u = 0; u < CPY_UNROLL; ++u) {
      size_t j = i + (size_t)u * nthreads;
      if (j < total4) lds2g_b128(dst + j, &stage[tid + u * CPY_BLOCK]);
    }
    async_wait0();  // stores drained LDS before we overwrite the stage buffer
  }
}

// Scalar tail for N*N % 4 != 0 (not hit for N=20000, kept for generality).
__global__ void copy_tail(const float* __restrict__ src, float* __restrict__ dst,
                          size_t start, size_t total) {
  size_t i = start + (size_t)blockIdx.x * blockDim.x + threadIdx.x;
  if (i < total) dst[i] = src[i];
}

// ---------------------------------------------------------------------------
// Kernel 2: new_h = zeros(N, D); new_h[idx] = h, in a single pass.
// One block per output row; block-uniform binary search over sorted int64 idx
// (compiler scalarizes: ~14 SMEM loads), then b128 row copy or zero fill.
// Every output element written exactly once.
// ---------------------------------------------------------------------------
__global__ __launch_bounds__(64) void scatter_rows(
    const float4* __restrict__ h, const long long* __restrict__ idx,
    float4* __restrict__ out, int n, int D4) {
  const int row = blockIdx.x;

  // lower_bound(idx, row)
  int lo = 0, hi = n;
  while (lo < hi) {
    int mid = (lo + hi) >> 1;
    if (idx[mid] < (long long)row) lo = mid + 1; else hi = mid;
  }
  const bool found = (lo < n) && (idx[lo] == (long long)row);

  const float4* srow = h + (size_t)lo * D4;
  float4* drow = out + (size_t)row * D4;
  const float4 zero = make_float4(0.f, 0.f, 0.f, 0.f);

  for (int c = threadIdx.x; c < D4; c += blockDim.x) {
    float4 v = zero;
    if (found) v = srow[c];   // guarded: no OOB read when row not kept
    drow[c] = v;
  }
}

// ---------------------------------------------------------------------------
// Host-side launcher. Inputs (setup_inputs order): g[N*N] f32, h[n*D] f32,
// pre_h[N*D] f32 (unused), idx[n] int64. d_out = [g (N*N) | new_h (N*D)].
// ---------------------------------------------------------------------------
extern "C" void kernel_launch(void* const* d_in, const int* in_sizes, int n_in,
                              void* d_out, int out_size, void* d_ws, size_t ws_size,
                              hipStream_t stream) {
  (void)n_in; (void)out_size; (void)d_ws; (void)ws_size;

  const float* g = (const float*)d_in[0];
  const float* h = (const float*)d_in[1];
  const long long* idx = (const long long*)d_in[3];

  const long long n = in_sizes[3];                 // kept rows
  const long long D = in_sizes[1] / n;             // feature dim (256)
  const long long N = in_sizes[2] / D;             // total rows (20000)

  float* out_g = (float*)d_out;
  float* out_h = out_g + (size_t)N * (size_t)N;

  const size_t totalF = (size_t)N * (size_t)N;
  const size_t total4 = totalF / 4;
  const size_t rem = totalF - total4 * 4;

  // Bulk g copy via async LDS DMA.
  copy_g_async<<<4096, CPY_BLOCK, 0, stream>>>(
      (const float4*)g, (float4*)out_g, total4);
  if (rem) {
    copy_tail<<<1, 64, 0, stream>>>(g, out_g, total4 * 4, totalF);
  }

  // Single-pass scatter + zero fill of new_h.
  scatter_rows<<<dim3((unsigned)N), 64, 0, stream>>>(
      (const float4*)h, idx, (float4*)out_h, (int)n, (int)(D / 4));
}